// QuantumLayer_61375082660332
// MI455X (gfx1250) — compile-verified
//
#include <hip/hip_runtime.h>
#include <math.h>

// CDNA5 / gfx1250, wave32.
// Quantum layer reduced to:  Y = ENC @ [Re(U)^T | Im(U)^T]  via V_WMMA_F32_16X16X4_F32,
// then probs + signed ds_bpermute butterfly (Walsh) for the 4 PauliZ expectations.

typedef __attribute__((ext_vector_type(2))) float v2f;
typedef __attribute__((ext_vector_type(8))) float v8f;

#define NQ 4
#define NL 2
#define PI4 0.78539816339744830962f

// ---------------------------------------------------------------------------
// Kernel 1: build the 16x16 complex unitary U of the variational circuit by
// evolving the 16 basis columns (one column per lane, register-resident).
//   W[k*16 + n]       = Re(U[n][k])   (B-matrix-ready layout)
//   W[256 + k*16 + n] = Im(U[n][k])
// ---------------------------------------------------------------------------
__global__ void build_unitary_kernel(const float* __restrict__ weights,
                                     float* __restrict__ W) {
    const int col = threadIdx.x & 15;
    float2 amp[16];
#pragma unroll
    for (int s = 0; s < 16; ++s) amp[s] = make_float2(0.f, 0.f);
    amp[col].x = 1.0f;

#pragma unroll
    for (int l = 0; l < NL; ++l) {
#pragma unroll
        for (int q = 0; q < NQ; ++q) {
            const int m = 1 << (3 - q);
            float th = weights[(l * NQ + q) * 2 + 0];
            float sy, cy;
            sincosf(0.5f * th, &sy, &cy);
#pragma unroll
            for (int s = 0; s < 16; ++s) {
                if (!(s & m)) {
                    float2 a0 = amp[s], a1 = amp[s | m];
                    amp[s]     = make_float2(cy * a0.x - sy * a1.x, cy * a0.y - sy * a1.y);
                    amp[s | m] = make_float2(sy * a0.x + cy * a1.x, sy * a0.y + cy * a1.y);
                }
            }
            float tz = weights[(l * NQ + q) * 2 + 1];
            float sz, cz;
            sincosf(0.5f * tz, &sz, &cz);
#pragma unroll
            for (int s = 0; s < 16; ++s) {
                float ssz = (s & m) ? sz : -sz;
                float2 a = amp[s];
                amp[s] = make_float2(cz * a.x - ssz * a.y, cz * a.y + ssz * a.x);
            }
        }
#pragma unroll
        for (int q = 0; q < NQ - 1; ++q) {
            const int cm = 1 << (3 - q);
            const int tm = 1 << (2 - q);
#pragma unroll
            for (int s = 0; s < 16; ++s) {
                if ((s & cm) && !(s & tm)) {
                    float2 t = amp[s];
                    amp[s] = amp[s | tm];
                    amp[s | tm] = t;
                }
            }
        }
    }

    if (threadIdx.x < 16) {
#pragma unroll
        for (int n = 0; n < 16; ++n) {
            W[col * 16 + n]       = amp[n].x;
            W[256 + col * 16 + n] = amp[n].y;
        }
    }
}

// ---------------------------------------------------------------------------
// Per-tile work: 16 samples.
//   A (16x4 f32, kk=0..3): A[M][K] = enc[sample M][basis K], M = lane%16,
//     K = v + 2*half + 4*kk.  Since (K>>2)==kk and (K&3)==2*half+v, the A
//     fragment is hi[kk] * (half ? lo[2+v] : lo[v]) — only two runtime selects.
//   B (4x16 f32): B[K][N] = Re/Im(U[N][K]); N = lane%16, same K mapping.
//   D (16x16 f32): lane holds N = lane%16, M = v + 8*half, v = 0..7.
// Then probs + 4-stage signed shfl_xor butterfly (Walsh over N): the Walsh
// coefficient at one-hot N = 1<<(3-q) equals <Z_q>.
// ---------------------------------------------------------------------------
__device__ __forceinline__ void compute_tile(int tile, int lane, int nn, int half,
                                             const float* __restrict__ x,
                                             const v2f br[4], const v2f bi[4],
                                             float ev[8]) {
    const int samp = (tile << 4) + nn;
    const float4 xv = ((const float4*)x)[samp];

    float c0, s0, c1, s1, c2, s2, c3, s3;
    __sincosf(xv.x * PI4, &s0, &c0);
    __sincosf(xv.y * PI4, &s1, &c1);
    __sincosf(xv.z * PI4, &s2, &c2);
    __sincosf(xv.w * PI4, &s3, &c3);

    // enc[j] = hi[j>>2] * lo[j&3]  (qubit0 = MSB of j)
    const float hi[4] = {c0 * c1, c0 * s1, s0 * c1, s0 * s1};
    const float l0 = c2 * c3, l1 = c2 * s3, l2 = s2 * c3, l3 = s2 * s3;
    const float loA = half ? l2 : l0;   // (K&3) = 2*half + 0
    const float loB = half ? l3 : l1;   // (K&3) = 2*half + 1

    v8f accR = {};
    v8f accI = {};
#pragma unroll
    for (int kk = 0; kk < 4; ++kk) {
        v2f a;
        a.x = hi[kk] * loA;
        a.y = hi[kk] * loB;
        accR = __builtin_amdgcn_wmma_f32_16x16x4_f32(
            false, a, false, br[kk], (short)0, accR, false, false);
        accI = __builtin_amdgcn_wmma_f32_16x16x4_f32(
            false, a, false, bi[kk], (short)0, accI, false, false);
    }

#pragma unroll
    for (int v = 0; v < 8; ++v)
        ev[v] = accR[v] * accR[v] + accI[v] * accI[v];

    // signed butterfly over the 4 bits of nn (stays within the 16-lane half)
#pragma unroll
    for (int b = 0; b < 4; ++b) {
        const int m = 1 << b;
#pragma unroll
        for (int v = 0; v < 8; ++v) {
            float other = __shfl_xor(ev[v], m, 32);
            ev[v] = (lane & m) ? (other - ev[v]) : (ev[v] + other);
        }
    }
}

__global__ __launch_bounds__(256) void qlayer_kernel(const float* __restrict__ x,
                                                     const float* __restrict__ W,
                                                     float* __restrict__ out, int batch) {
    const int lane   = threadIdx.x & 31;
    const int half   = lane >> 4;
    const int nn     = lane & 15;
    const int wave   = blockIdx.x * (blockDim.x >> 5) + (threadIdx.x >> 5);
    const int nwaves = gridDim.x * (blockDim.x >> 5);

    // Constant B matrices, kept in VGPRs for the whole kernel.
    v2f br[4], bi[4];
#pragma unroll
    for (int kk = 0; kk < 4; ++kk) {
        const int k0 = 4 * kk + 2 * half;
        br[kk].x = W[k0 * 16 + nn];
        br[kk].y = W[(k0 + 1) * 16 + nn];
        bi[kk].x = W[256 + k0 * 16 + nn];
        bi[kk].y = W[256 + (k0 + 1) * 16 + nn];
    }

    const int onehot = (__popc(nn) == 1);
    const int q = (nn == 8) ? 0 : (nn == 4) ? 1 : (nn == 2) ? 2 : 3;

    const int ntiles = batch >> 4;
    const int npairs = ntiles >> 1;

    // Branch-free dual-tile body: two independent WMMA/butterfly chains.
    for (int p = wave; p < npairs; p += nwaves) {
        const int t0 = 2 * p;
        const int t1 = 2 * p + 1;
        float evA[8], evB[8];
        compute_tile(t0, lane, nn, half, x, br, bi, evA);
        compute_tile(t1, lane, nn, half, x, br, bi, evB);

        if (onehot) {
#pragma unroll
            for (int v = 0; v < 8; ++v) {
                out[((t0 << 4) + v + 8 * half) * 4 + q] = evA[v];
                out[((t1 << 4) + v + 8 * half) * 4 + q] = evB[v];
            }
        }
    }

    // Odd-tile tail (batch=524288 -> never taken, kept for generality).
    if ((ntiles & 1) && wave == 0) {
        float evT[8];
        compute_tile(ntiles - 1, lane, nn, half, x, br, bi, evT);
        if (onehot) {
#pragma unroll
            for (int v = 0; v < 8; ++v)
                out[(((ntiles - 1) << 4) + v + 8 * half) * 4 + q] = evT[v];
        }
    }
}

extern "C" void kernel_launch(void* const* d_in, const int* in_sizes, int n_in,
                              void* d_out, int out_size, void* d_ws, size_t ws_size,
                              hipStream_t stream) {
    const float* x       = (const float*)d_in[0];   // (B, 4) f32
    const float* weights = (const float*)d_in[1];   // (2, 4, 2) f32
    float* out = (float*)d_out;                     // (B, 4) f32
    float* W   = (float*)d_ws;                      // 512 floats: Re|Im of U
    const int batch = in_sizes[0] / NQ;

    build_unitary_kernel<<<1, 32, 0, stream>>>(weights, W);
    qlayer_kernel<<<1024, 256, 0, stream>>>(x, W, out, batch);
}